// BoxSelector_30099130811138
// MI455X (gfx1250) — compile-verified
//
#include <hip/hip_runtime.h>

#define B_IMG 128
#define N_PRI 32768
#define C_CH  6
#define TOPK  400
#define KEEPK 200
#define NMS_T 0.5f
#define CONF_T 0.5f
#define BIGV 1000000000.0f

typedef float v2f __attribute__((ext_vector_type(2)));
typedef float v8f __attribute__((ext_vector_type(8)));
typedef unsigned int v4u __attribute__((ext_vector_type(4)));
typedef int v4i __attribute__((ext_vector_type(4)));
typedef int v8i __attribute__((ext_vector_type(8)));

#if __has_builtin(__builtin_amdgcn_tensor_load_to_lds)
#define HAVE_TDM 1
#else
#define HAVE_TDM 0
#endif

__device__ __forceinline__ unsigned f2ascend(float f) {
  unsigned u = __float_as_uint(f);
  return (u & 0x80000000u) ? ~u : (u | 0x80000000u);
}

__device__ __forceinline__ void wait_tensorcnt0() {
#if __has_builtin(__builtin_amdgcn_s_wait_tensorcnt)
  __builtin_amdgcn_s_wait_tensorcnt(0);
#else
  asm volatile("s_wait_tensorcnt 0" ::: "memory");
#endif
}

// ---------------------------------------------------------------------------
// Kernel 1: per image -- stage conf pairs via TDM, build (desc-score|idx) keys,
// full 32768 bitonic sort in LDS, emit top-400 (score, label, decoded box).
// ---------------------------------------------------------------------------
__global__ __launch_bounds__(1024) void topk_kernel(
    const float* __restrict__ pred, const float* __restrict__ priors,
    float* __restrict__ top_s, float* __restrict__ labs,
    float* __restrict__ boxesW) {
  const int b = blockIdx.x;
  const int tid = threadIdx.x;

  extern __shared__ unsigned long long smem[];
  unsigned long long* keys = smem;                 // 32768 * 8B = 256 KB
  float* stg = (float*)(smem + N_PRI);             // 2048 * 2 * 4B = 16 KB

  // ---- build keys, chunk by chunk (16 chunks of 2048 priors) ----
  for (int c = 0; c < 16; ++c) {
#if HAVE_TDM
    if (tid < 32) {  // one wave issues the TDM DMA
      unsigned long long gaddr = (unsigned long long)(uintptr_t)(
          pred + ((size_t)b * N_PRI + (size_t)c * 2048) * C_CH + 4);
      unsigned lds_addr = (unsigned)(uintptr_t)stg;
      v4u g0;
      g0[0] = 1u;                                   // count=1 (valid D#)
      g0[1] = lds_addr;                             // lds_addr[31:0]
      g0[2] = (unsigned)gaddr;                      // global_addr[31:0]
      g0[3] = (unsigned)((gaddr >> 32) & 0x01FFFFFFull) | 0x80000000u; // type=2
      v8i g1;
      g1[0] = 0x00020000;        // data_size = 2 (4 bytes)
      g1[1] = 0x00020000;        // tensor_dim0 = 2 (low16 at bits 63:48)
      g1[2] = (int)0x08000000;   // tensor_dim1 = 2048 (low16 at bits 95:80)
      g1[3] = 0x00020000;        // tile_dim0 = 2
      g1[4] = 0x00000800;        // tile_dim1 = 2048
      g1[5] = 6;                 // tensor_dim0_stride = 6
      g1[6] = 0;
      g1[7] = 0;
      v4i z4 = {0, 0, 0, 0};
#if defined(__clang_major__) && (__clang_major__ >= 23)
      v8i z8 = {0, 0, 0, 0, 0, 0, 0, 0};
      __builtin_amdgcn_tensor_load_to_lds(g0, g1, z4, z4, z8, 0);
#else
      __builtin_amdgcn_tensor_load_to_lds(g0, g1, z4, z4, 0);
#endif
      wait_tensorcnt0();
    }
    __syncthreads();
#else
    for (int e = tid; e < 2048; e += 1024) {
      int gi = c * 2048 + e;
      stg[2 * e + 0] = pred[((size_t)b * N_PRI + gi) * C_CH + 4];
      stg[2 * e + 1] = pred[((size_t)b * N_PRI + gi) * C_CH + 5];
    }
    __syncthreads();
#endif
    for (int e = tid; e < 2048; e += 1024) {
      int gi = c * 2048 + e;
      float c0 = stg[2 * e + 0];
      float c1 = stg[2 * e + 1];
      float sc = fmaxf(c0, c1);
      float m = (sc > CONF_T) ? sc : -BIGV;
      unsigned u = f2ascend(m);
      // ascending-sort key => descending score, ties -> lower index first
      keys[gi] = ((unsigned long long)(~u) << 32) | (unsigned)gi;
    }
    __syncthreads();
  }

  // ---- bitonic sort 32768 u64 keys in LDS (pair-index form) ----
  for (unsigned k = 2; k <= (unsigned)N_PRI; k <<= 1) {
    for (unsigned j = k >> 1; j > 0; j >>= 1) {
      __syncthreads();
      for (unsigned t = tid; t < (unsigned)(N_PRI / 2); t += 1024) {
        unsigned i = ((t & ~(j - 1)) << 1) | (t & (j - 1));
        unsigned l = i | j;
        unsigned long long a = keys[i];
        unsigned long long b2 = keys[l];
        bool asc = ((i & k) == 0);
        if (asc ? (a > b2) : (a < b2)) {
          keys[i] = b2;
          keys[l] = a;
        }
      }
    }
  }
  __syncthreads();

  // ---- emit top-400: recompute masked score/label, decode box ----
  if (tid < TOPK) {
    unsigned long long key = keys[tid];
    int i = (int)(unsigned)(key & 0xFFFFFFFFull);
    const float* p = pred + ((size_t)b * N_PRI + i) * C_CH;
    float c0 = p[4], c1 = p[5];
    float sc = fmaxf(c0, c1);
    float m = (sc > CONF_T) ? sc : -BIGV;
    const float* pr = priors + (size_t)i * 4;
    float cx = pr[0] + p[0] * 0.1f * pr[2];
    float cy = pr[1] + p[1] * 0.1f * pr[3];
    float w = pr[2] * expf(p[2] * 0.2f);
    float h = pr[3] * expf(p[3] * 0.2f);
    size_t o = (size_t)b * TOPK + tid;
    top_s[o] = m;
    labs[o] = (c1 > c0) ? 1.0f : 0.0f;
    boxesW[o * 4 + 0] = cx - w * 0.5f;
    boxesW[o * 4 + 1] = cy - h * 0.5f;
    boxesW[o * 4 + 2] = cx + w * 0.5f;
    boxesW[o * 4 + 3] = cy + h * 0.5f;
  }
}

// ---------------------------------------------------------------------------
// Kernel 2: per image -- WMMA-tiled IoU adjacency bitmask, sequential greedy
// NMS over bitmask, stable ascending bitonic argsort, write 200x6 output.
// ---------------------------------------------------------------------------
__global__ __launch_bounds__(512) void nms_kernel(
    const float* __restrict__ top_s, const float* __restrict__ labs,
    const float* __restrict__ boxesW, float* __restrict__ out) {
  const int b = blockIdx.x;
  const int tid = threadIdx.x;

  __shared__ float box_s[TOPK * 4];
  __shared__ float score_s[TOPK];
  __shared__ float label_s[TOPK];
  __shared__ float area_s[TOPK];
  __shared__ unsigned adj_s[TOPK * 16];  // 512-bit row per candidate
  __shared__ unsigned kw[16];            // keep bitmask (13 words used)
  __shared__ unsigned long long keys2[512];

  for (int i = tid; i < TOPK * 16; i += 512) adj_s[i] = 0u;
  if (tid < 16) kw[tid] = 0u;
  __syncthreads();

  if (tid < TOPK) {
    size_t o = (size_t)b * TOPK + tid;
    float s = top_s[o];
    score_s[tid] = s;
    label_s[tid] = labs[o];
    float x1 = boxesW[o * 4 + 0], y1 = boxesW[o * 4 + 1];
    float x2 = boxesW[o * 4 + 2], y2 = boxesW[o * 4 + 3];
    box_s[tid * 4 + 0] = x1;
    box_s[tid * 4 + 1] = y1;
    box_s[tid * 4 + 2] = x2;
    box_s[tid * 4 + 3] = y2;
    area_s[tid] = (x2 - x1) * (y2 - y1);
    if (s > CONF_T) atomicOr(&kw[tid >> 5], 1u << (tid & 31));
  }
  __syncthreads();

  // ---- adjacency: 25x25 tiles of 16x16; WMMA computes area_i + area_j ----
  const int wave = tid >> 5;
  const int lane = tid & 31;
  for (int t5 = wave; t5 < 625; t5 += 16) {  // wave-uniform loop, EXEC all-1s
    int ti = t5 / 25;
    int tj = t5 - ti * 25;
    float a0 = 0.0f, a1 = 0.0f, b0 = 0.0f, b1 = 0.0f;
    if (lane < 16) {
      a0 = area_s[ti * 16 + lane];  // A[:,K=0] = area_i
      a1 = 1.0f;                    // A[:,K=1] = 1
      b0 = 1.0f;                    // B[K=0,:] = 1
      b1 = area_s[tj * 16 + lane];  // B[K=1,:] = area_j
    }
    v2f Af = {a0, a1};
    v2f Bf = {b0, b1};
    v8f Cf = {};
    // D[m][n] = area_i[m] + area_j[n]  (16x16x4 f32 matrix op)
    v8f Df = __builtin_amdgcn_wmma_f32_16x16x4_f32(false, Af, false, Bf,
                                                   (short)0, Cf, false, false);
    int nn = lane & 15;
    int hi = lane >> 4;
#pragma unroll
    for (int r = 0; r < 8; ++r) {
      int i = ti * 16 + hi * 8 + r;  // C/D layout: lanes 16-31 hold M=8..15
      int j = tj * 16 + nn;
      if (j > i) {
        const float* bi = &box_s[i * 4];
        const float* bj = &box_s[j * 4];
        float iw = fminf(bi[2], bj[2]) - fmaxf(bi[0], bj[0]);
        float ih = fminf(bi[3], bj[3]) - fmaxf(bi[1], bj[1]);
        iw = fmaxf(iw, 0.0f);
        ih = fmaxf(ih, 0.0f);
        float inter = iw * ih;
        // iou > T  <=>  inter > T * (area_i + area_j - inter + 1e-12)
        // (denominator strictly positive: priors in (0,1), w/h = p*exp(.) > 0)
        float denom = Df[r] - inter + 1e-12f;
        if (inter > NMS_T * denom)
          atomicOr(&adj_s[i * 16 + (j >> 5)], 1u << (j & 31));
      }
    }
  }
  __syncthreads();

  // ---- sequential greedy NMS over bitmask (matches reference fori_loop) ----
  if (tid == 0) {
    volatile unsigned* vk = kw;
    for (int i = 0; i < TOPK; ++i) {
      if ((vk[i >> 5] >> (i & 31)) & 1u) {
        const unsigned* arow = &adj_s[i * 16];
        for (int w = 0; w < 13; ++w) vk[w] &= ~arow[w];
      }
    }
  }
  __syncthreads();

  // ---- stable ascending argsort of (keep ? score : BIG), 512-wide bitonic --
  {
    unsigned long long key;
    if (tid < TOPK) {
      bool kp = (kw[tid >> 5] >> (tid & 31)) & 1u;
      float sk = kp ? score_s[tid] : BIGV;
      key = ((unsigned long long)f2ascend(sk) << 32) | (unsigned)tid;
    } else {
      key = ~0ull;  // padding sorts after all real keys
    }
    keys2[tid] = key;
  }
  for (int k = 2; k <= 512; k <<= 1) {
    for (int j = k >> 1; j > 0; j >>= 1) {
      __syncthreads();
      int p2 = tid ^ j;
      if (p2 > tid) {
        unsigned long long a = keys2[tid];
        unsigned long long b2 = keys2[p2];
        bool asc = ((tid & k) == 0);
        if (asc ? (a > b2) : (a < b2)) {
          keys2[tid] = b2;
          keys2[p2] = a;
        }
      }
    }
  }
  __syncthreads();

  // ---- write 200 rows: [label, score, x1, y1, x2, y2] or zeros ----
  if (tid < KEEPK) {
    unsigned long long key = keys2[tid];
    int ord = (int)(unsigned)(key & 0xFFFFFFFFull);
    bool kp = (kw[ord >> 5] >> (ord & 31)) & 1u;
    size_t base = ((size_t)b * KEEPK + tid) * 6;
    if (kp) {
      out[base + 0] = label_s[ord];
      out[base + 1] = score_s[ord];
      out[base + 2] = box_s[ord * 4 + 0];
      out[base + 3] = box_s[ord * 4 + 1];
      out[base + 4] = box_s[ord * 4 + 2];
      out[base + 5] = box_s[ord * 4 + 3];
    } else {
      for (int c = 0; c < 6; ++c) out[base + c] = 0.0f;
    }
  }
}

extern "C" void kernel_launch(void* const* d_in, const int* in_sizes, int n_in,
                              void* d_out, int out_size, void* d_ws,
                              size_t ws_size, hipStream_t stream) {
  const float* pred = (const float*)d_in[0];    // (128, 32768, 6) f32
  const float* priors = (const float*)d_in[1];  // (32768, 4) f32
  float* ws = (float*)d_ws;
  float* tops = ws;                         // B*400
  float* labsw = ws + (size_t)B_IMG * TOPK; // B*400
  float* boxesW = ws + (size_t)2 * B_IMG * TOPK;  // B*400*4
  float* out = (float*)d_out;               // B*200*6

  size_t shmem = (size_t)N_PRI * 8 + (size_t)2048 * 2 * 4;  // keys + staging
  topk_kernel<<<B_IMG, 1024, shmem, stream>>>(pred, priors, tops, labsw,
                                              boxesW);
  nms_kernel<<<B_IMG, 512, 0, stream>>>(tops, labsw, boxesW, out);
}